// GAT_29781303231106
// MI455X (gfx1250) — compile-verified
//
#include <hip/hip_runtime.h>

// ---------------------------------------------------------------------------
// GAT (2 layers, heads=1) for MI455X / gfx1250 (wave32, WMMA)
//   layer: h = act(A)@W ; as=h@a_s ; ad=h@a_d ;
//          e = leaky_relu(as[src]+ad[dst]) ; segment-softmax by dst ;
//          out[dst] += alpha * h[src]
// GEMMs use V_WMMA_F32_16X16X4_F32 (exact fp32 path on the matrix pipe).
// B operand staged in LDS pair-interleaved so each fragment is ONE
// ds_load_b64 into an even VGPR pair (no repack movs in the inner loop).
// Edge phase is HBM/L2-bandwidth bound; uses f32 atomics (L2-resident dsts).
// ---------------------------------------------------------------------------

typedef __attribute__((ext_vector_type(2))) float v2f;
typedef __attribute__((ext_vector_type(8))) float v8f;

#define LRELU_NEG 0.2f

__device__ __forceinline__ int enc_f32(float f) {
  // order-preserving float -> signed int encoding (involution)
  int i = __float_as_int(f);
  return i >= 0 ? i : (i ^ 0x7fffffff);
}
__device__ __forceinline__ float dec_f32(int e) {
  int i = e >= 0 ? e : (e ^ 0x7fffffff);
  return __int_as_float(i);
}

// --------------------------- fill helpers ----------------------------------
__global__ void fill_f32_k(float* __restrict__ p, float v, int n) {
  int i = blockIdx.x * blockDim.x + threadIdx.x;
  if (i < n) p[i] = v;
}
__global__ void fill_i32_k(int* __restrict__ p, int v, int n) {
  int i = blockIdx.x * blockDim.x + threadIdx.x;
  if (i < n) p[i] = v;
}

// --------------------- fused GEMM + attention dots -------------------------
// A: [nrows, 128] (optionally relu(A + bias_in) when FUSE_RELU)
// W: [128, NOUT] row-major; staged in LDS pair-interleaved over K:
//    sWp[p*NOUT + n] = { W[2p][n], W[2p+1][n] }
// One block = 256 threads = 8 waves; each wave computes a 16 x NOUT strip via
// v_wmma_f32_16x16x4_f32, then reduces h.a_s / h.a_d in-register + shfl_xor.
template <int NOUT, bool FUSE_RELU>
__global__ void __launch_bounds__(256)
gat_gemm_alpha(const float* __restrict__ A, const float* __restrict__ W,
               const float* __restrict__ bias_in,
               const float* __restrict__ avec_s, const float* __restrict__ avec_d,
               float* __restrict__ H, float* __restrict__ alpha_s,
               float* __restrict__ alpha_d, int nrows) {
  constexpr int KDIM = 128;
  constexpr int NT = NOUT / 16;

  __shared__ v2f sWp[(KDIM / 2) * NOUT];  // 64KB (NOUT=128) / 32KB (NOUT=64)
  for (int i = threadIdx.x; i < KDIM * NOUT; i += 256) {
    int k = i / NOUT, n = i - k * NOUT;
    sWp[(k >> 1) * NOUT + n][k & 1] = W[i];
  }
  __syncthreads();

  const int lane = threadIdx.x & 31;
  const int wv   = threadIdx.x >> 5;
  const int hf   = lane >> 4;   // half-wave: 0 -> K,K+1 ; 1 -> K+2,K+3
  const int l16  = lane & 15;
  const int row0 = blockIdx.x * 128 + wv * 16;

  int ra = row0 + l16;
  if (ra > nrows - 1) ra = nrows - 1;  // clamp: WMMA needs EXEC all-ones
  const float* __restrict__ arow = A + (size_t)ra * KDIM;
  const int koff = 2 * hf;

  v8f acc[NT] = {};
  #pragma unroll 4
  for (int k = 0; k < KDIM; k += 4) {
    // A fragment (16x4 f32): lane<16 holds A[r][k..k+1], lane>=16 A[r][k+2..k+3]
    float ax = arow[k + koff];
    float ay = arow[k + koff + 1];
    if (FUSE_RELU) {
      ax += bias_in[k + koff];     ax = ax > 0.f ? ax : 0.f;
      ay += bias_in[k + koff + 1]; ay = ay > 0.f ? ay : 0.f;
    }
    v2f a; a.x = ax; a.y = ay;
    const v2f* __restrict__ brow = sWp + (k / 2 + hf) * NOUT + l16;
    #pragma unroll
    for (int n = 0; n < NT; ++n) {
      // B fragment: {W[k+koff][col], W[k+koff+1][col]} = one ds_load_b64
      v2f b = brow[n * 16];
      acc[n] = __builtin_amdgcn_wmma_f32_16x16x4_f32(
          false, a, false, b, (short)0, acc[n], false, false);
    }
  }

  // epilogue 1: store H (C layout: VGPR r -> row r + 8*hf, col n*16 + l16)
  #pragma unroll
  for (int n = 0; n < NT; ++n) {
    #pragma unroll
    for (int r = 0; r < 8; ++r) {
      int row = row0 + r + 8 * hf;
      if (row < nrows) H[(size_t)row * NOUT + n * 16 + l16] = acc[n][r];
    }
  }

  // epilogue 2: alpha_s/alpha_d = h . a_vec   (reduce over NOUT cols)
  float rs[8], rd[8];
  #pragma unroll
  for (int r = 0; r < 8; ++r) { rs[r] = 0.f; rd[r] = 0.f; }
  #pragma unroll
  for (int n = 0; n < NT; ++n) {
    float cs = avec_s[n * 16 + l16];
    float cd = avec_d[n * 16 + l16];
    #pragma unroll
    for (int r = 0; r < 8; ++r) {
      rs[r] += acc[n][r] * cs;
      rd[r] += acc[n][r] * cd;
    }
  }
  // xor-reduce within each 16-lane half (bits 0..3 only; halves independent)
  #pragma unroll
  for (int r = 0; r < 8; ++r) {
    #pragma unroll
    for (int off = 1; off < 16; off <<= 1) {
      rs[r] += __shfl_xor(rs[r], off, 32);
      rd[r] += __shfl_xor(rd[r], off, 32);
    }
  }
  #pragma unroll
  for (int r = 0; r < 8; ++r) {
    int row = row0 + r + 8 * hf;
    if (l16 == r && row < nrows) {
      alpha_s[row] = rs[r];
      alpha_d[row] = rd[r];
    }
  }
}

// ------------------------- edge pass 1: segment max ------------------------
__global__ void edge_max_k(const int* __restrict__ ei, int E, int Nn,
                           const float* __restrict__ as_,
                           const float* __restrict__ ad_,
                           int* __restrict__ menc) {
  int t = blockIdx.x * blockDim.x + threadIdx.x;
  if (t >= E + Nn) return;
  int src = (t < E) ? ei[t]     : (t - E);   // self-loop tail
  int dst = (t < E) ? ei[E + t] : (t - E);
  float s = as_[src] + ad_[dst];
  s = s > 0.f ? s : LRELU_NEG * s;
  atomicMax(menc + dst, enc_f32(s));
}

// ------------------------- edge pass 2: exp-sum ----------------------------
__global__ void edge_sum_k(const int* __restrict__ ei, int E, int Nn,
                           const float* __restrict__ as_,
                           const float* __restrict__ ad_,
                           const int* __restrict__ menc,
                           float* __restrict__ denom) {
  int t = blockIdx.x * blockDim.x + threadIdx.x;
  if (t >= E + Nn) return;
  int src = (t < E) ? ei[t]     : (t - E);
  int dst = (t < E) ? ei[E + t] : (t - E);
  float s = as_[src] + ad_[dst];
  s = s > 0.f ? s : LRELU_NEG * s;
  atomicAdd(denom + dst, __expf(s - dec_f32(menc[dst])));
}

// ---------------- edge pass 3: weighted scatter-aggregate ------------------
// One wave per edge; each lane moves F/32 contiguous floats (vector load).
template <int F>
__global__ void __launch_bounds__(256)
edge_agg_k(const int* __restrict__ ei, int E, int Nn,
           const float* __restrict__ as_, const float* __restrict__ ad_,
           const int* __restrict__ menc, const float* __restrict__ denom,
           const float* __restrict__ Hsrc, float* __restrict__ Out) {
  int gw   = (blockIdx.x * blockDim.x + threadIdx.x) >> 5;
  int lane = threadIdx.x & 31;
  if (gw >= E + Nn) return;
  int src = (gw < E) ? ei[gw]     : (gw - E);
  int dst = (gw < E) ? ei[E + gw] : (gw - E);
  float s = as_[src] + ad_[dst];
  s = s > 0.f ? s : LRELU_NEG * s;
  float w = __expf(s - dec_f32(menc[dst])) / (denom[dst] + 1e-16f);

  constexpr int VPL = F / 32;  // 4 (layer1) or 2 (layer2)
  const float* __restrict__ hp = Hsrc + (size_t)src * F + lane * VPL;
  float* __restrict__ op = Out + (size_t)dst * F + lane * VPL;
  if (VPL == 4) {
    float4 hv = *(const float4*)hp;  // global_load_b128
    atomicAdd(op + 0, w * hv.x);
    atomicAdd(op + 1, w * hv.y);
    atomicAdd(op + 2, w * hv.z);
    atomicAdd(op + 3, w * hv.w);
  } else {
    float2 hv = *(const float2*)hp;  // global_load_b64
    atomicAdd(op + 0, w * hv.x);
    atomicAdd(op + 1, w * hv.y);
  }
}

// --------------------------- final bias add --------------------------------
__global__ void add_bias_k(float* __restrict__ out, const float* __restrict__ b,
                           int n, int C) {
  int i = blockIdx.x * blockDim.x + threadIdx.x;
  if (i < n) out[i] += b[i % C];
}

// ---------------------------------------------------------------------------
extern "C" void kernel_launch(void* const* d_in, const int* in_sizes, int n_in,
                              void* d_out, int out_size, void* d_ws,
                              size_t ws_size, hipStream_t stream) {
  const float* x      = (const float*)d_in[0];
  const int*   ei     = (const int*)  d_in[1];
  const float* W1     = (const float*)d_in[2];
  const float* a_src1 = (const float*)d_in[3];
  const float* a_dst1 = (const float*)d_in[4];
  const float* b1     = (const float*)d_in[5];
  const float* W2     = (const float*)d_in[6];
  const float* a_src2 = (const float*)d_in[7];
  const float* a_dst2 = (const float*)d_in[8];
  const float* b2     = (const float*)d_in[9];
  float* out = (float*)d_out;

  const int N = in_sizes[0] / 128;  // 100000
  const int E = in_sizes[1] / 2;    // 1600000
  const int Etot = E + N;

  // workspace layout (floats)
  float* h1   = (float*)d_ws;            // N*128 (reused as h2 after layer1)
  float* out1 = h1 + (size_t)N * 128;    // N*128
  float* as_  = out1 + (size_t)N * 128;  // N
  float* ad_  = as_ + N;                 // N
  int*   menc = (int*)(ad_ + N);         // N
  float* den  = (float*)(menc + N);      // N
  float* h2   = h1;                      // alias: h1 dead after layer-1 agg

  const int gGemm  = (N + 127) / 128;
  const int gNode  = (N + 255) / 256;
  const int gEdge  = (Etot + 255) / 256;
  const int gAgg   = (Etot + 7) / 8;  // 8 waves/block, wave per edge
  const int gFeat1 = (N * 128 + 255) / 256;
  const int gOut   = (N * 64 + 255) / 256;

  // ------------------------------ layer 1 ------------------------------
  gat_gemm_alpha<128, false><<<gGemm, 256, 0, stream>>>(
      x, W1, nullptr, a_src1, a_dst1, h1, as_, ad_, N);
  fill_i32_k<<<gNode, 256, 0, stream>>>(menc, (int)0x80000000, N);
  fill_f32_k<<<gNode, 256, 0, stream>>>(den, 0.f, N);
  fill_f32_k<<<gFeat1, 256, 0, stream>>>(out1, 0.f, N * 128);
  edge_max_k<<<gEdge, 256, 0, stream>>>(ei, E, N, as_, ad_, menc);
  edge_sum_k<<<gEdge, 256, 0, stream>>>(ei, E, N, as_, ad_, menc, den);
  edge_agg_k<128><<<gAgg, 256, 0, stream>>>(ei, E, N, as_, ad_, menc, den, h1, out1);

  // ------------------------------ layer 2 ------------------------------
  // h2 = relu(out1 + b1) @ W2 fused into the GEMM A-operand load
  gat_gemm_alpha<64, true><<<gGemm, 256, 0, stream>>>(
      out1, W2, b1, a_src2, a_dst2, h2, as_, ad_, N);
  fill_i32_k<<<gNode, 256, 0, stream>>>(menc, (int)0x80000000, N);
  fill_f32_k<<<gNode, 256, 0, stream>>>(den, 0.f, N);
  fill_f32_k<<<gOut, 256, 0, stream>>>(out, 0.f, N * 64);
  edge_max_k<<<gEdge, 256, 0, stream>>>(ei, E, N, as_, ad_, menc);
  edge_sum_k<<<gEdge, 256, 0, stream>>>(ei, E, N, as_, ad_, menc, den);
  edge_agg_k<64><<<gAgg, 256, 0, stream>>>(ei, E, N, as_, ad_, menc, den, h2, out);
  add_bias_k<<<gOut, 256, 0, stream>>>(out, b2, N * 64, 64);
}